// VoxelToPixel_40295383171316
// MI455X (gfx1250) — compile-verified
//
#include <hip/hip_runtime.h>
#include <stdint.h>

#define BB 4
#define HH 240
#define WW 320
#define CC 64
#define HW (HH * WW)
#define DUMP (BB * HW)

typedef __attribute__((ext_vector_type(2))) float v2f;
typedef __attribute__((ext_vector_type(8))) float v8f;

// ---------------------------------------------------------------------------
// CDNA5 async global->LDS copy (ASYNCcnt path, CDNA5 ISA)
// ---------------------------------------------------------------------------
__device__ __forceinline__ void async_ld_b128(const float* src, unsigned lds_byte_off) {
    asm volatile("global_load_async_to_lds_b128 %0, %1, off"
                 :: "v"(lds_byte_off), "v"(src) : "memory");
}
__device__ __forceinline__ void wait_async0() {
    asm volatile("s_wait_asynccnt 0" ::: "memory");
}

// ---------------------------------------------------------------------------
// Init: depth buffer = bits(1e30f), winner buffer = N
// ---------------------------------------------------------------------------
__global__ void vtp_init_kernel(unsigned* __restrict__ depthbuf,
                                int* __restrict__ winbuf, int N) {
    int i = blockIdx.x * blockDim.x + threadIdx.x;
    if (i <= DUMP) {
        depthbuf[i] = __float_as_uint(1e30f);
        winbuf[i]   = N;
    }
}

// ---------------------------------------------------------------------------
// Projection of 32 points per wave via two V_WMMA_F32_16X16X4_F32.
// A (16x4 f32): lanes 0-15 hold {K0,K1}={x,y}; lanes 16-31 hold {K2,K3}={z,1}.
// B (4x16 f32): VGPR0 = rows K=0 (lanes 0-15) / K=2 (lanes 16-31), VGPR1 = K=1/K=3.
// D (16x16 f32): VGPR j, lane L -> D[j + 8*(L>>4)][L&15]; unscrambled via LDS.
// ---------------------------------------------------------------------------
__device__ __forceinline__ void project32(const int* __restrict__ coords,
                                          const float* __restrict__ P, int N,
                                          int p_base, int lane, float* wl,
                                          int& bidx, int& p,
                                          float& px, float& py, float& pz) {
    p = p_base + lane;
    int pc = (p < N) ? p : (N - 1);
    const int4 cc = ((const int4*)coords)[pc];   // (b, z, y, x)
    bidx = cc.x;
    float xf = (float)cc.w, yf = (float)cc.z, zf = (float)cc.y;

    // marshal A for the two 16-point WMMAs
    float zlow = __shfl(zf, lane & 15, 32);          // z of points 0..15
    float xhi  = __shfl(xf, (lane & 15) + 16, 32);   // x of points 16..31
    float yhi  = __shfl(yf, (lane & 15) + 16, 32);   // y of points 16..31
    bool lo = (lane < 16);

    v2f a1, a2, bm;
    a1.x = lo ? xf  : zlow;  a1.y = lo ? yf  : 1.0f;
    a2.x = lo ? xhi : zf;    a2.y = lo ? yhi : 1.0f;

    // B[k][n] = P[n][k] (n<3), zero-padded to 16 columns
    int n = lane & 15;
    float b0 = 0.0f, b1 = 0.0f;
    if (n < 3) {
        b0 = P[n * 4 + (lo ? 0 : 2)];
        b1 = P[n * 4 + (lo ? 1 : 3)];
    }
    bm.x = b0; bm.y = b1;

    v8f acc = {0.f, 0.f, 0.f, 0.f, 0.f, 0.f, 0.f, 0.f};
    v8f d1 = __builtin_amdgcn_wmma_f32_16x16x4_f32(false, a1, false, bm,
                                                   (short)0, acc, false, false);
    v8f d2 = __builtin_amdgcn_wmma_f32_16x16x4_f32(false, a2, false, bm,
                                                   (short)0, acc, false, false);

    // scatter D through LDS so lane m owns point m's row
    int col = lane & 15;
    int rh  = (lane >> 4) * 8;
#pragma unroll
    for (int j = 0; j < 8; ++j) wl[(rh + j) * 16 + col]      = d1[j];
#pragma unroll
    for (int j = 0; j < 8; ++j) wl[(16 + rh + j) * 16 + col] = d2[j];
    __syncthreads();
    px = wl[lane * 16 + 0];
    py = wl[lane * 16 + 1];
    pz = wl[lane * 16 + 2];
}

// mode 0: atomic float-min (as u32 bits) z-buffer.  mode 1: winner index min.
__global__ void vtp_proj_kernel(const int* __restrict__ coords,
                                const float* __restrict__ P, int N,
                                unsigned* __restrict__ depthbuf,
                                int* __restrict__ winbuf, int mode) {
    __shared__ float lds[8 * 512];                    // 8 waves * 32 rows * 16 cols
    int lane = threadIdx.x & 31;
    int wid  = threadIdx.x >> 5;
    int p_base = (blockIdx.x * 8 + wid) * 32;

    int bidx, p;
    float px, py, pz;
    project32(coords, P, N, p_base, lane, &lds[wid * 512], bidx, p, px, py, pz);

    float depth = pz;
    int u = (int)floorf(px / depth);
    int v = (int)floorf(py / depth);
    bool valid = (depth > 1e-6f) && (u >= 0) && (u < WW) &&
                 (v >= 0) && (v < HH) && (p < N);
    if (valid) {
        int lin = bidx * HW + v * WW + u;
        if (mode == 0) {
            // positive floats: uint ordering == float ordering
            atomicMin(depthbuf + lin, __float_as_uint(depth));
        } else {
            if (__float_as_uint(depth) == depthbuf[lin])
                atomicMin(winbuf + lin, p);
        }
    }
}

// ---------------------------------------------------------------------------
// Gather: 64 pixels per block. Async-stage winner feature rows (256B each)
// into a padded LDS tile, then store channel-major with coalesced writes.
// ---------------------------------------------------------------------------
#define PITCH 68   // 64 + 4 floats pad: keeps 16B alignment, breaks bank collisions

__global__ void vtp_gather_kernel(const float* __restrict__ features,
                                  const unsigned* __restrict__ depthbuf,
                                  const int* __restrict__ winbuf, int N,
                                  float* __restrict__ out,
                                  float* __restrict__ invd) {
    __shared__ float fbuf[64 * PITCH];
    __shared__ int   winb[64];

    int t  = threadIdx.x;
    int p0 = blockIdx.x * 64;

    if (t < 64) {
        int p = p0 + t;
        int w = winbuf[p];
        winb[t] = w;
        float d = __uint_as_float(depthbuf[p]);
        invd[p] = (w < N) ? (1.0f / fmaxf(d, 1e-30f)) : 0.0f;
    }
    __syncthreads();

    int pix = t >> 2;          // 64 pixels, 4 threads each
    int q   = t & 3;           // each thread covers 16 channels
    int w   = winb[pix];
    if (w < N) {
        const float* src = features + (size_t)w * CC + q * 16;
        unsigned dst = (unsigned)(size_t)(&fbuf[pix * PITCH + q * 16]);
#pragma unroll
        for (int i = 0; i < 4; ++i)
            async_ld_b128(src + i * 4, dst + i * 16);
    } else {
#pragma unroll
        for (int i = 0; i < 16; ++i)
            fbuf[pix * PITCH + q * 16 + i] = 0.0f;
    }
    wait_async0();
    __syncthreads();

    // transposed store: lanes sweep hw -> fully coalesced per channel
    int pix2 = t & 63;
    int c0   = t >> 6;
    int p    = p0 + pix2;
    int b    = p / HW;
    int hw   = p - b * HW;
    size_t obase = (size_t)b * CC * HW + hw;
#pragma unroll
    for (int c = c0; c < CC; c += 4)
        out[obase + (size_t)c * HW] = fbuf[pix2 * PITCH + c];
}

// ---------------------------------------------------------------------------
extern "C" void kernel_launch(void* const* d_in, const int* in_sizes, int n_in,
                              void* d_out, int out_size, void* d_ws, size_t ws_size,
                              hipStream_t stream) {
    const float* features = (const float*)d_in[0];
    const int*   coords   = (const int*)d_in[1];
    const float* projmat  = (const float*)d_in[2];
    int N = in_sizes[0] / CC;

    unsigned* depthbuf = (unsigned*)d_ws;
    int*      winbuf   = (int*)(depthbuf + (DUMP + 1));

    float* out  = (float*)d_out;
    float* invd = out + (size_t)BB * CC * HW;

    // 1) init z-buffer / winner buffer
    vtp_init_kernel<<<(DUMP + 256) / 256, 256, 0, stream>>>(depthbuf, winbuf, N);

    // 2) WMMA projection + float z-buffer min
    int projBlocks = (N + 255) / 256;
    vtp_proj_kernel<<<projBlocks, 256, 0, stream>>>(coords, projmat, N,
                                                    depthbuf, winbuf, 0);
    // 3) winner (min point index among depth ties)
    vtp_proj_kernel<<<projBlocks, 256, 0, stream>>>(coords, projmat, N,
                                                    depthbuf, winbuf, 1);
    // 4) async-staged gather + transposed coalesced store
    vtp_gather_kernel<<<DUMP / 64, 256, 0, stream>>>(features, depthbuf, winbuf,
                                                     N, out, invd);
}